// DistanceEncoder_88751204204749
// MI455X (gfx1250) — compile-verified
//
#include <hip/hip_runtime.h>
#include <cstdint>
#include <cstddef>

// DistanceEncoder for MI455X (gfx1250, wave32) — WMMA formulation, v3.
//
// haversine dis = 0.5*acos(dot(u_ent, u_q)) with 3D unit vectors
// u = (cos(lat)cos(lon), cos(lat)sin(lon), sin(lat)).
// Per batch b: biased dots via V_WMMA_F32_16X16X4_F32 with K=4 operands
//   A[s] = (x, y, z, 1)        B[:,l] = (x_l, y_l, z_l, bias_l)
//   D = A*B + 0  (SRC2 = inline 0)  =>  dot3 + (masked ? -1e4 : 0)
// -min_l dis = -0.5*acos(max_l D[s,l]); all-masked rows (max < -100) -> -2.0.
//
// A/B lane fragments are pre-swizzled in LDS so the inner loop is exactly
// ds_load_b64 -> v_wmma -> v_max. Input staging uses the gfx1250 async
// global->LDS path (ASYNCcnt).

typedef float v2f __attribute__((ext_vector_type(2)));
typedef float v8f __attribute__((ext_vector_type(8)));

#define HPI_F 1.57079632679489662f   // deg2rad(90)
#define PI_F  3.14159265358979323f   // deg2rad(180)

namespace {

constexpr int L  = 256;   // max_locs       (N dimension)
constexpr int S  = 128;   // samples per qa (M dimension)
constexpr int NT = 256;   // 8 wave32s per block
constexpr float MASK_BIAS = -1.0e4f;

// Low 32 bits of a generic LDS pointer == LDS byte offset (ISA §10.2).
__device__ __forceinline__ uint32_t lds_byte_off(const void* p) {
    return (uint32_t)(uintptr_t)p;
}
__device__ __forceinline__ void async_g2l_b64(uint32_t lds, const void* g) {
    asm volatile("global_load_async_to_lds_b64 %0, %1, off"
                 :: "v"(lds), "v"((uint64_t)(uintptr_t)g) : "memory");
}
__device__ __forceinline__ void async_g2l_b32(uint32_t lds, const void* g) {
    asm volatile("global_load_async_to_lds_b32 %0, %1, off"
                 :: "v"(lds), "v"((uint64_t)(uintptr_t)g) : "memory");
}
__device__ __forceinline__ void wait_async0() {
    asm volatile("s_wait_asynccnt 0x0" ::: "memory");
}

} // anonymous namespace

__global__ __launch_bounds__(NT)
void DistanceEncoder_88751204204749_kernel(const float* __restrict__ q,
                                           const uint8_t* __restrict__ mask,
                                           const float* __restrict__ ent,
                                           float* __restrict__ out) {
    __shared__ float2   qstage[L];      // raw question lat/long pairs
    __shared__ uint32_t mstage[L / 4];  // raw mask bytes
    __shared__ float2   estage[S];      // raw entity lat/long pairs
    __shared__ float2   btab[2 * L];    // B fragments: [l]=(x,y)  [L+l]=(z,bias)
    __shared__ float2   atab[2 * S];    // A fragments: [s]=(x,y)  [S+s]=(z,1)
    __shared__ float    rowmax[S];      // per-sample max biased dot

    const int b = blockIdx.x;
    const int t = threadIdx.x;

    // ---- Phase 1: async global -> LDS staging (ASYNCcnt path) ----
    async_g2l_b64(lds_byte_off(&qstage[t]), (const float2*)q + (size_t)b * L + t);
    if (t < S)
        async_g2l_b64(lds_byte_off(&estage[t]), (const float2*)ent + (size_t)b * S + t);
    if (t < L / 4)
        async_g2l_b32(lds_byte_off(&mstage[t]),
                      (const uint32_t*)(const void*)(mask + (size_t)b * L) + t);
    wait_async0();
    __syncthreads();

    // ---- Phase 2: unit vectors, pre-swizzled into WMMA lane fragments ----
    {
        const float2 qp = qstage[t];
        const float ph = qp.x * HPI_F, lm = qp.y * PI_F;
        const float sp = __sinf(ph), cp = __cosf(ph);
        const float sl = __sinf(lm), cl = __cosf(lm);
        const uint32_t mword = mstage[t >> 2];
        const bool valid = ((mword >> ((t & 3) * 8)) & 0xFFu) != 0u;
        btab[t]     = make_float2(cp * cl, cp * sl);               // (x, y)   K=0,1
        btab[L + t] = make_float2(sp, valid ? 0.0f : MASK_BIAS);   // (z, bias) K=2,3
        if (t < S) {
            const float2 ep = estage[t];
            const float phc = ep.x * HPI_F, lmc = ep.y * PI_F;
            const float spc = __sinf(phc), cpc = __cosf(phc);
            const float slc = __sinf(lmc), clc = __cosf(lmc);
            atab[t]     = make_float2(cpc * clc, cpc * slc);       // (x, y)  K=0,1
            atab[S + t] = make_float2(spc, 1.0f);                  // (z, 1)  K=2,3
        }
    }
    __syncthreads();

    // ---- Phase 3: per-wave M-tile GEMM via V_WMMA_F32_16X16X4_F32 ----
    // 16x4 f32 A / 4x16 f32 B: lanes 0-15 carry K=0,1; lanes 16-31 carry K=2,3.
    // C/D 16x16 f32: lane n, VGPR r = row r (lanes 0-15) / row r+8 (lanes 16-31).
    const int wv   = t >> 5;      // wave id -> M-tile (rows wv*16..wv*16+15)
    const int lane = t & 31;
    const int half = lane >> 4;   // fragment half: K=0,1 vs K=2,3
    const int sub  = lane & 15;   // row (A) / column (B,C,D) within tile

    const float2 av = atab[half * S + wv * 16 + sub];
    const v2f a = {av.x, av.y};
    const float2* bp = &btab[half * L + sub];   // per-lane base; j*16 folds into
                                                // the ds_load_b64 immediate offset
    const v8f zero = {};                        // SRC2 -> inline constant 0
    v8f acc;
#pragma unroll
    for (int j = 0; j < 16; ++j) {              // N-tiles: columns j*16..j*16+15
        const float2 bv = bp[j * 16];
        const v2f bb = {bv.x, bv.y};
        const v8f d = __builtin_amdgcn_wmma_f32_16x16x4_f32(
            /*neg_a=*/false, a, /*neg_b=*/false, bb,
            /*c_mod=*/(short)0, zero, /*reuse_a=*/false, /*reuse_b=*/false);
        if (j == 0) {
            acc = d;
        } else {
#pragma unroll
            for (int r = 0; r < 8; ++r) acc[r] = fmaxf(acc[r], d[r]);
        }
    }

    // ---- Phase 4: max across the 16 lanes of each column group ----
#pragma unroll
    for (int r = 0; r < 8; ++r) {
        float v = acc[r];
        v = fmaxf(v, __shfl_xor(v, 1));
        v = fmaxf(v, __shfl_xor(v, 2));
        v = fmaxf(v, __shfl_xor(v, 4));
        v = fmaxf(v, __shfl_xor(v, 8));
        if (sub == 0) rowmax[wv * 16 + half * 8 + r] = v;  // row r (+8 high half)
    }
    __syncthreads();

    // ---- Phase 5: finalize: -0.5*acos(maxdot), sentinel for all-masked ----
    if (t < S) {
        const float m = rowmax[t];
        out[(size_t)b * S + t] =
            (m < -100.0f) ? -2.0f
                          : -0.5f * acosf(fminf(fmaxf(m, -1.0f), 1.0f));
    }
}

extern "C" void kernel_launch(void* const* d_in, const int* in_sizes, int n_in,
                              void* d_out, int out_size, void* d_ws, size_t ws_size,
                              hipStream_t stream) {
    (void)n_in; (void)d_ws; (void)ws_size; (void)out_size;
    const float*   q    = (const float*)d_in[0];    // [B, L, 2] f32
    const uint8_t* mask = (const uint8_t*)d_in[1];  // [B, L] bool (1 byte/elem)
    const float*   ent  = (const float*)d_in[2];    // [B, S, 2] f32
    float*         out  = (float*)d_out;            // [B, S] f32

    const int B = in_sizes[1] / L;                  // mask has B*L elements
    DistanceEncoder_88751204204749_kernel<<<dim3(B), dim3(NT), 0, stream>>>(
        q, mask, ent, out);
}